// GCN_GRU_Model_58729382806112
// MI455X (gfx1250) — compile-verified
//
#include <hip/hip_runtime.h>
#include <hip/hip_bf16.h>

#define NODES_H 16   // GCN/GRU hidden
#define OUT_CH  32

typedef __attribute__((ext_vector_type(2))) float v2f;
typedef __attribute__((ext_vector_type(8))) float v8f;

// ---------------------------------------------------------------------------
// Degree: in-degree at dst (float), one thread per edge.
__global__ void k_degree(const int* __restrict__ dst, float* __restrict__ deg, int E) {
    int e = blockIdx.x * blockDim.x + threadIdx.x;
    if (e >= E) return;
    atomicAdd(&deg[dst[e]], 1.0f);
}

// dinv = rsqrt(deg + 1)  (in place)
__global__ void k_dinv(float* __restrict__ deg, int n) {
    int i = blockIdx.x * blockDim.x + threadIdx.x;
    if (i >= n) return;
    deg[i] = rsqrtf(deg[i] + 1.0f);
}

// ---------------------------------------------------------------------------
// Dense GEMM  H[out] = X[nrows,K] @ W[K,16]  via V_WMMA_F32_16X16X4_F32.
// One wave per 16-row tile. nrows must be a multiple of 16 (200000 = 12500*16).
// A frag (16x4 f32): lanes 0-15 hold M=0..15 with K={k,k+1}; lanes 16-31 K={k+2,k+3}.
// B frag (4x16 f32): lane half selects K pair, lane&15 = output column N.
// C/D (16x16 f32, 8 VGPRs): VGPR i -> M=i (lane<16) / M=8+i (lane>=16), N=lane&15.
template <int K>
__global__ void k_gemm_wmma(const float* __restrict__ X, const float* __restrict__ W,
                            float* __restrict__ Hout, int nrows) {
    int wave = blockIdx.x * (blockDim.x >> 5) + (threadIdx.x >> 5);
    int lane = threadIdx.x & 31;
    int row0 = wave * 16;
    if (row0 >= nrows) return;              // wave-uniform: EXEC stays all-ones for WMMA

    int m     = lane & 15;                  // tile row (A) / tile col (B,C,D)
    int khalf = lane >> 4;                  // K-pair select within each 4-wide K step
    const float* arow = X + (size_t)(row0 + m) * K + 2 * khalf;

    v8f acc = {0.f, 0.f, 0.f, 0.f, 0.f, 0.f, 0.f, 0.f};
#pragma unroll
    for (int k = 0; k < K; k += 4) {
        v2f a, b;
        a[0] = arow[k + 0];
        a[1] = arow[k + 1];
        int kb = k + 2 * khalf;
        b[0] = W[(kb + 0) * NODES_H + m];
        b[1] = W[(kb + 1) * NODES_H + m];
        acc = __builtin_amdgcn_wmma_f32_16x16x4_f32(false, a, false, b,
                                                    (short)0, acc, false, false);
    }

    int n     = lane & 15;
    int mbase = (lane >> 4) * 8;
    float* orow = Hout + (size_t)(row0 + mbase) * NODES_H + n;
#pragma unroll
    for (int i = 0; i < 8; ++i)
        orow[(size_t)i * NODES_H] = acc[i];
}

// ---------------------------------------------------------------------------
// Edge aggregation: agg[dst] += h[src] * dinv[src] * dinv[dst], 16 channels/edge.
__global__ void k_edge_agg(const int* __restrict__ src, const int* __restrict__ dst,
                           const float* __restrict__ h, const float* __restrict__ dinv,
                           float* __restrict__ agg, int E) {
    int e = blockIdx.x * blockDim.x + threadIdx.x;
    if (e >= E) return;
    int s = src[e], d = dst[e];
    float w = dinv[s] * dinv[d];
    const float4* hp = (const float4*)(h + (size_t)s * NODES_H);
    float* ap = agg + (size_t)d * NODES_H;
#pragma unroll
    for (int q = 0; q < 4; ++q) {
        float4 v = hp[q];
        atomicAdd(ap + 4 * q + 0, v.x * w);
        atomicAdd(ap + 4 * q + 1, v.y * w);
        atomicAdd(ap + 4 * q + 2, v.z * w);
        atomicAdd(ap + 4 * q + 3, v.w * w);
    }
}

// out = relu(agg + h*dinv^2 + b)   one thread per (node, channel)
__global__ void k_post(const float* __restrict__ agg, const float* __restrict__ h,
                       const float* __restrict__ dinv, const float* __restrict__ b,
                       float* __restrict__ out, int n16) {
    int t = blockIdx.x * blockDim.x + threadIdx.x;
    if (t >= n16) return;
    int node = t >> 4;
    int c    = t & 15;
    float d  = dinv[node];
    float v  = agg[t] + h[t] * d * d + b[c];
    out[t] = fmaxf(v, 0.0f);
}

// ---------------------------------------------------------------------------
// Fused GRU (seq_len=1, h0=0) + final Linear. One thread per node; weights in LDS.
__global__ void k_gru_fc(const float* __restrict__ h, const float* __restrict__ w_ih,
                         const float* __restrict__ b_ih, const float* __restrict__ b_hh,
                         const float* __restrict__ Wfc, const float* __restrict__ bfc,
                         float* __restrict__ out, int n) {
    __shared__ float s_wih[3 * NODES_H * NODES_H];   // [48,16]
    __shared__ float s_bih[3 * NODES_H];
    __shared__ float s_bhh[3 * NODES_H];
    __shared__ float s_wfc[OUT_CH * NODES_H];        // [32,16]
    __shared__ float s_bfc[OUT_CH];

    for (int t = threadIdx.x; t < 3 * NODES_H * NODES_H; t += blockDim.x) s_wih[t] = w_ih[t];
    for (int t = threadIdx.x; t < 3 * NODES_H; t += blockDim.x) { s_bih[t] = b_ih[t]; s_bhh[t] = b_hh[t]; }
    for (int t = threadIdx.x; t < OUT_CH * NODES_H; t += blockDim.x) s_wfc[t] = Wfc[t];
    for (int t = threadIdx.x; t < OUT_CH; t += blockDim.x) s_bfc[t] = bfc[t];
    __syncthreads();

    int i = blockIdx.x * blockDim.x + threadIdx.x;
    if (i >= n) return;

    float hv[NODES_H];
    const float4* hp = (const float4*)(h + (size_t)i * NODES_H);
#pragma unroll
    for (int q = 0; q < 4; ++q) {
        float4 v = hp[q];
        hv[4 * q + 0] = v.x; hv[4 * q + 1] = v.y; hv[4 * q + 2] = v.z; hv[4 * q + 3] = v.w;
    }

    float gi[3 * NODES_H];
#pragma unroll
    for (int j = 0; j < 3 * NODES_H; ++j) {
        float acc = s_bih[j];
        const float* wr = s_wih + j * NODES_H;
#pragma unroll
        for (int k = 0; k < NODES_H; ++k) acc = fmaf(hv[k], wr[k], acc);
        gi[j] = acc;
    }

    float hs[NODES_H];
#pragma unroll
    for (int c = 0; c < NODES_H; ++c) {
        // gh = h0 @ w_hh^T + b_hh = b_hh  (h0 == 0)
        float r  = 1.0f / (1.0f + __expf(-(gi[c] + s_bhh[c])));
        float z  = 1.0f / (1.0f + __expf(-(gi[NODES_H + c] + s_bhh[NODES_H + c])));
        float nn = tanhf(gi[2 * NODES_H + c] + r * s_bhh[2 * NODES_H + c]);
        hs[c] = (1.0f - z) * nn;                 // + z*h0 == 0
    }

    float* orow = out + (size_t)i * OUT_CH;
#pragma unroll
    for (int o = 0; o < OUT_CH; ++o) {
        float acc = s_bfc[o];
        const float* wr = s_wfc + o * NODES_H;
#pragma unroll
        for (int k = 0; k < NODES_H; ++k) acc = fmaf(hs[k], wr[k], acc);
        orow[o] = acc;
    }
}

// ---------------------------------------------------------------------------
extern "C" void kernel_launch(void* const* d_in, const int* in_sizes, int n_in,
                              void* d_out, int out_size, void* d_ws, size_t ws_size,
                              hipStream_t stream) {
    const float* x    = (const float*)d_in[0];
    const int*   ei   = (const int*)d_in[1];
    // d_in[2] = batch (unused: single graph, GRU h0 = 0)
    const float* W1   = (const float*)d_in[3];
    const float* b1   = (const float*)d_in[4];
    const float* W2   = (const float*)d_in[5];
    const float* b2   = (const float*)d_in[6];
    const float* w_ih = (const float*)d_in[7];
    // d_in[8] = w_hh (unused: multiplies h0 == 0)
    const float* b_ih = (const float*)d_in[9];
    const float* b_hh = (const float*)d_in[10];
    const float* Wfc  = (const float*)d_in[11];
    const float* bfc  = (const float*)d_in[12];
    float*       out  = (float*)d_out;

    const int N = in_sizes[2];            // nodes (batch has N entries)
    const int E = in_sizes[1] / 2;        // edges
    const int* src = ei;
    const int* dst = ei + E;

    // Workspace: dinv[N] | h[N*16] | agg[N*16] | h2[N*16]   (~39.2 MB for N=200k)
    char*  ws    = (char*)d_ws;
    float* dinv  = (float*)ws;
    float* hbuf  = (float*)(ws + (size_t)N * 4);
    float* aggb  = hbuf + (size_t)N * NODES_H;
    float* hbuf2 = aggb + (size_t)N * NODES_H;

    const int TB = 256;
    dim3 eb((E + TB - 1) / TB), nb((N + TB - 1) / TB), cb(((size_t)N * NODES_H + TB - 1) / TB);
    int waves = (N + 15) / 16;
    dim3 gb((waves + 7) / 8);             // 8 waves (256 threads) per block

    // ---- degrees / norms
    hipMemsetAsync(dinv, 0, (size_t)N * 4, stream);
    k_degree<<<eb, TB, 0, stream>>>(dst, dinv, E);
    k_dinv<<<nb, TB, 0, stream>>>(dinv, N);

    // ---- GCN layer 1: h = X @ W1 ; agg ; relu(agg + h*dinv^2 + b1)
    hipMemsetAsync(aggb, 0, (size_t)N * NODES_H * 4, stream);
    k_gemm_wmma<128><<<gb, TB, 0, stream>>>(x, W1, hbuf, N);
    k_edge_agg<<<eb, TB, 0, stream>>>(src, dst, hbuf, dinv, aggb, E);
    k_post<<<cb, TB, 0, stream>>>(aggb, hbuf, dinv, b1, hbuf2, N * NODES_H);

    // ---- GCN layer 2: h = h' @ W2 ; agg ; relu(... + b2)
    hipMemsetAsync(aggb, 0, (size_t)N * NODES_H * 4, stream);
    k_gemm_wmma<16><<<gb, TB, 0, stream>>>(hbuf2, W2, hbuf, N);
    k_edge_agg<<<eb, TB, 0, stream>>>(src, dst, hbuf, dinv, aggb, E);
    k_post<<<cb, TB, 0, stream>>>(aggb, hbuf, dinv, b2, hbuf2, N * NODES_H);

    // ---- GRU (seq=1, h0=0) + FC
    k_gru_fc<<<nb, TB, 0, stream>>>(hbuf2, w_ih, b_ih, b_hh, Wfc, bfc, out, N);
}